// Net_14336600834600
// MI455X (gfx1250) — compile-verified
//
#include <hip/hip_runtime.h>
#include <math.h>

// ---------------------------------------------------------------------------
// Sizes (static, from the reference)
// ---------------------------------------------------------------------------
#define BGRAPH   128
#define NPG      32768           // nodes per graph
#define K1N      328             // ceil(0.01*32768)
#define K2N      33              // ceil(0.1*328)
#define K3N      4               // ceil(0.1*33)
#define M1       (BGRAPH * K1N)  // 41984
#define M2       (BGRAPH * K2N)  // 4224

typedef float v2f __attribute__((ext_vector_type(2)));
typedef float v8f __attribute__((ext_vector_type(8)));

// Monotone float->uint key (descending float order == descending uint order)
__device__ __forceinline__ unsigned int f2key(float f) {
  unsigned int u = __float_as_uint(f);
  return (u & 0x80000000u) ? ~u : (u | 0x80000000u);
}
__device__ __forceinline__ float key2f(unsigned int k) {
  unsigned int u = (k & 0x80000000u) ? (k ^ 0x80000000u) : ~k;
  return __uint_as_float(u);
}

// ---------------------------------------------------------------------------
// Kernel 1: pool1. One block per graph (1024 threads).
//   - scoring pass reads x as 5x float4 per 4 rows (80B groups, 16B aligned)
//     -> pure b128 traffic on the 84 MB dominant stream
//   - all 32768 keys live in LDS (128 KB of the 320 KB WGP LDS)
//   - 4-pass radix select of the K1N-th largest key (threshold)
//   - compact >=T candidates, bitonic-sort 512 composites (key<<32 | ~idx)
//     -> exact lax.top_k order (desc value, asc index on ties)
//   - emit xp1[g*328+r][0..7] = x[idx]*tanh(dot/||w1||), rows padded to 8
// ---------------------------------------------------------------------------
__global__ void pool1_kernel(const float* __restrict__ x,
                             const float* __restrict__ w1,
                             float* __restrict__ xp1) {
  extern __shared__ unsigned char smem[];
  unsigned int*       keys = (unsigned int*)smem;                   // 32768
  unsigned int*       hist = (unsigned int*)(smem + 131072);        // 256
  unsigned int*       ctrl = hist + 256;                            // 8 words
  unsigned long long* cand = (unsigned long long*)(ctrl + 8);       // 512

  const int g   = blockIdx.x;
  const int tid = threadIdx.x;
  const int nt  = blockDim.x;

  float w0 = w1[0], w1v = w1[1], w2v = w1[2], w3v = w1[3], w4v = w1[4];
  const float inv_norm = rsqrtf(w0*w0 + w1v*w1v + w2v*w2v + w3v*w3v + w4v*w4v);

  const float* xg = x + (size_t)g * NPG * 5;

  // ---- score pass: 4 rows per group, 5 aligned float4 loads per group ----
  for (int gi = tid; gi < NPG / 4; gi += nt) {
    const float4* p = (const float4*)(xg + (size_t)gi * 20);
    __builtin_prefetch(xg + (size_t)(gi + nt) * 20, 0, 0);
    float4 q0 = p[0], q1 = p[1], q2 = p[2], q3 = p[3], q4 = p[4];
    float d0 = q0.x*w0 + q0.y*w1v + q0.z*w2v + q0.w*w3v + q1.x*w4v;
    float d1 = q1.y*w0 + q1.z*w1v + q1.w*w2v + q2.x*w3v + q2.y*w4v;
    float d2 = q2.z*w0 + q2.w*w1v + q3.x*w2v + q3.y*w3v + q3.z*w4v;
    float d3 = q3.w*w0 + q4.x*w1v + q4.y*w2v + q4.z*w3v + q4.w*w4v;
    uint4 kq;
    kq.x = f2key(d0); kq.y = f2key(d1); kq.z = f2key(d2); kq.w = f2key(d3);
    *(uint4*)(keys + 4 * gi) = kq;   // one ds_store_b128
  }
  if (tid == 0) { ctrl[2] = 0u; ctrl[3] = K1N; }   // prefix, need
  __syncthreads();

  // ---- 4-pass radix select (MSB first) for the K1N-th largest key ----
  for (int shift = 24; shift >= 0; shift -= 8) {
    for (int b = tid; b < 256; b += nt) hist[b] = 0u;
    __syncthreads();
    const unsigned int prefix = ctrl[2];
    const unsigned int maskHi = (shift == 24) ? 0u : (0xFFFFFFFFu << (shift + 8));
    for (int i = tid; i < NPG; i += nt) {
      unsigned int k = keys[i];
      if ((k & maskHi) == prefix) atomicAdd(&hist[(k >> shift) & 0xFFu], 1u);
    }
    __syncthreads();
    if (tid == 0) {
      unsigned int need = ctrl[3], cum = 0u; int sel = 0;
      for (int b = 255; b >= 0; --b) {
        if (cum + hist[b] >= need) { sel = b; break; }
        cum += hist[b];
      }
      ctrl[2] = prefix | ((unsigned int)sel << shift);
      ctrl[3] = need - cum;
    }
    __syncthreads();
  }
  const unsigned int T = ctrl[2];

  // ---- compact candidates: all > T, then == T ----
  if (tid < 2) ctrl[tid] = 0u;
  for (int i = tid; i < 512; i += nt) cand[i] = 0ull;
  __syncthreads();
  for (int i = tid; i < NPG; i += nt) {
    unsigned int k = keys[i];
    if (k > T) {
      unsigned int p = atomicAdd(&ctrl[0], 1u);
      if (p < 512u) cand[p] = ((unsigned long long)k << 32) | (unsigned int)(~i);
    }
  }
  __syncthreads();
  const unsigned int cgt = ctrl[0];
  for (int i = tid; i < NPG; i += nt) {
    if (keys[i] == T) {
      unsigned int p = cgt + atomicAdd(&ctrl[1], 1u);
      if (p < 512u) cand[p] = ((unsigned long long)T << 32) | (unsigned int)(~i);
    }
  }
  __syncthreads();

  // ---- bitonic ascending sort of 512 composites ----
  for (int kk = 2; kk <= 512; kk <<= 1) {
    for (int jj = kk >> 1; jj > 0; jj >>= 1) {
      for (int i = tid; i < 512; i += nt) {
        int l = i ^ jj;
        if (l > i) {
          unsigned long long a = cand[i], b = cand[l];
          bool up = ((i & kk) == 0);
          if (up ? (a > b) : (a < b)) { cand[i] = b; cand[l] = a; }
        }
      }
      __syncthreads();
    }
  }

  // ---- emit top K1N rows (rank r = cand[511-r]), padded to 8 floats ----
  for (int r = tid; r < K1N; r += nt) {
    unsigned long long c = cand[511 - r];
    unsigned int key = (unsigned int)(c >> 32);
    int idx = (int)(~(unsigned int)c);
    float val = tanhf(key2f(key) * inv_norm);
    const float* xr = xg + (size_t)idx * 5;
    float4 o01 = make_float4(xr[0]*val, xr[1]*val, xr[2]*val, xr[3]*val);
    float4 o2  = make_float4(xr[4]*val, 0.f, 0.f, 0.f);
    float* o = xp1 + (size_t)(g * K1N + r) * 8;   // 32B-aligned rows
    *(float4*)(o)     = o01;
    *(float4*)(o + 4) = o2;
  }
}

// ---------------------------------------------------------------------------
// Kernel 2: RNNCell  y1 = tanh(xp1 @ W_ih^T + b_ih + b_hh)   [41984 x 64]
// WMMA f32 16x16x4, K padded 5->8. W_ih staged zero-padded in LDS so the
// B-fragment load is an unconditional 8B LDS read (no EXEC-divergent branch
// between WMMAs). One wave per 16x16 tile; 10496 tiles / 8 waves = 1312
// blocks exact -> EXEC all-ones at every WMMA.
// ---------------------------------------------------------------------------
__global__ void rnn_gemm_wmma(const float* __restrict__ A,    // [M1 x 8]
                              const float* __restrict__ Wih,  // [64 x 5]
                              const float* __restrict__ bih,
                              const float* __restrict__ bhh,
                              float* __restrict__ Y) {        // [M1 x 64]
  __shared__ float wp[64][8];   // W_ih^T padded: wp[n][k] = (k<5) ? Wih[n][k] : 0
  __shared__ float bb[64];
  for (int i = threadIdx.x; i < 64 * 8; i += blockDim.x) {
    const int n = i >> 3, k = i & 7;
    wp[n][k] = (k < 5) ? Wih[n * 5 + k] : 0.f;
  }
  for (int i = threadIdx.x; i < 64; i += blockDim.x) bb[i] = bih[i] + bhh[i];
  __syncthreads();

  const int lane  = threadIdx.x & 31;
  const int wave  = threadIdx.x >> 5;
  const int tile  = blockIdx.x * 8 + wave;
  const int tm    = tile >> 2;        // 64/16 = 4 n-tiles
  const int tn    = tile & 3;
  const int row   = tm * 16 + (lane & 15);
  const int col   = tn * 16 + (lane & 15);
  const int khalf = (lane >> 4) * 2;  // lanes 0-15 -> K{0,1}; 16-31 -> K{2,3}

  v8f c = {};
#pragma unroll
  for (int kc = 0; kc < 2; ++kc) {
    const int kb = kc * 4 + khalf;
    v2f a = *(const v2f*)(A + (size_t)row * 8 + kb);   // global_load_b64
    v2f b = *(const v2f*)(&wp[col][kb]);               // ds_load_b64
    c = __builtin_amdgcn_wmma_f32_16x16x4_f32(false, a, false, b,
                                              (short)0, c, false, false);
  }
  const float bias = bb[col];
  const int mbase = tm * 16 + ((lane >> 4) ? 8 : 0);
#pragma unroll
  for (int v = 0; v < 8; ++v) {
    Y[(size_t)(mbase + v) * 64 + col] = tanhf(c[v] + bias);
  }
}

// ---------------------------------------------------------------------------
// Kernel 3: pool2. One block (256 thr) per graph: 328 scores -> top 33.
// Dots and gathers are float4-vectorized (rows are 256B, 16B-aligned).
// ---------------------------------------------------------------------------
__global__ void pool2_kernel(const float* __restrict__ y1,   // [M1 x 64]
                             const float* __restrict__ w2,   // [64]
                             float* __restrict__ xp2) {      // [M2 x 64]
  __shared__ unsigned long long cand[512];
  __shared__ float wsh[64];
  const int g = blockIdx.x, tid = threadIdx.x, nt = blockDim.x;

  if (tid < 64) wsh[tid] = w2[tid];
  for (int i = tid; i < 512; i += nt) cand[i] = 0ull;
  __syncthreads();

  float nrm = 0.f;
#pragma unroll
  for (int f = 0; f < 64; ++f) nrm += wsh[f] * wsh[f];
  const float inv_norm = rsqrtf(nrm);

  const float* yg = y1 + (size_t)g * K1N * 64;
  const float4* wsh4 = (const float4*)wsh;
  for (int i = tid; i < K1N; i += nt) {
    const float4* r4 = (const float4*)(yg + (size_t)i * 64);
    float d = 0.f;
#pragma unroll
    for (int f = 0; f < 16; ++f) {
      float4 q = r4[f], ww = wsh4[f];
      d += q.x*ww.x + q.y*ww.y + q.z*ww.z + q.w*ww.w;
    }
    cand[i] = ((unsigned long long)f2key(d) << 32) | (unsigned int)(~i);
  }
  __syncthreads();

  for (int kk = 2; kk <= 512; kk <<= 1) {
    for (int jj = kk >> 1; jj > 0; jj >>= 1) {
      for (int i = tid; i < 512; i += nt) {
        int l = i ^ jj;
        if (l > i) {
          unsigned long long a = cand[i], b = cand[l];
          bool up = ((i & kk) == 0);
          if (up ? (a > b) : (a < b)) { cand[i] = b; cand[l] = a; }
        }
      }
      __syncthreads();
    }
  }

  // gather + scale, float4 granularity: 33 rows x 16 quads
  for (int e = tid; e < K2N * 16; e += nt) {
    const int r = e >> 4, q = e & 15;
    unsigned long long c = cand[511 - r];
    unsigned int key = (unsigned int)(c >> 32);
    int idx = (int)(~(unsigned int)c);
    float val = tanhf(key2f(key) * inv_norm);
    float4 v = *(const float4*)(yg + (size_t)idx * 64 + q * 4);
    float4 o = make_float4(v.x*val, v.y*val, v.z*val, v.w*val);
    *(float4*)(xp2 + (size_t)(g * K2N + r) * 64 + q * 4) = o;
  }
}

// ---------------------------------------------------------------------------
// Kernel 4: y2 = relu(xp2 @ W2^T + b2)  [4224 x 32], WMMA f32 16x16x4, K=64.
// A/B fragments as aligned 8B loads. 528 tiles / 8 waves = 66 blocks exact.
// ---------------------------------------------------------------------------
__global__ void gemm2_wmma(const float* __restrict__ A,    // [M2 x 64]
                           const float* __restrict__ W2m,  // [32 x 64]
                           const float* __restrict__ b2,
                           float* __restrict__ Y) {        // [M2 x 32]
  const int lane  = threadIdx.x & 31;
  const int wave  = threadIdx.x >> 5;
  const int tile  = blockIdx.x * 8 + wave;
  const int tm    = tile >> 1;        // 32/16 = 2 n-tiles
  const int tn    = tile & 1;
  const int row   = tm * 16 + (lane & 15);
  const int col   = tn * 16 + (lane & 15);
  const int khalf = (lane >> 4) * 2;

  v8f c = {};
#pragma unroll
  for (int kc = 0; kc < 16; ++kc) {
    const int kb = kc * 4 + khalf;
    v2f a = *(const v2f*)(A   + (size_t)row * 64 + kb);
    v2f b = *(const v2f*)(W2m + (size_t)col * 64 + kb);
    c = __builtin_amdgcn_wmma_f32_16x16x4_f32(false, a, false, b,
                                              (short)0, c, false, false);
  }
  const float bias = b2[col];
  const int mbase = tm * 16 + ((lane >> 4) ? 8 : 0);
#pragma unroll
  for (int v = 0; v < 8; ++v) {
    Y[(size_t)(mbase + v) * 32 + col] = fmaxf(c[v] + bias, 0.f);
  }
}

// ---------------------------------------------------------------------------
// Kernel 5: pool3 (33 -> top 4) fused with final linear [32 -> 8].
// One block (64 thr) per graph.
// ---------------------------------------------------------------------------
__global__ void pool3_final_kernel(const float* __restrict__ y2,  // [M2 x 32]
                                   const float* __restrict__ w3,  // [32]
                                   const float* __restrict__ W3m, // [8 x 32]
                                   const float* __restrict__ b3,
                                   float* __restrict__ out) {     // [B*4 x 8]
  __shared__ unsigned long long cand[64];
  __shared__ float wsh[32];
  const int g = blockIdx.x, tid = threadIdx.x;

  if (tid < 32) wsh[tid] = w3[tid];
  cand[tid] = 0ull;
  __syncthreads();

  float nrm = 0.f;
#pragma unroll
  for (int f = 0; f < 32; ++f) nrm += wsh[f] * wsh[f];
  const float inv_norm = rsqrtf(nrm);

  const float* yg = y2 + (size_t)g * K2N * 32;
  if (tid < K2N) {
    const float4* r4 = (const float4*)(yg + (size_t)tid * 32);
    const float4* w4 = (const float4*)wsh;
    float d = 0.f;
#pragma unroll
    for (int f = 0; f < 8; ++f) {
      float4 q = r4[f], ww = w4[f];
      d += q.x*ww.x + q.y*ww.y + q.z*ww.z + q.w*ww.w;
    }
    cand[tid] = ((unsigned long long)f2key(d) << 32) | (unsigned int)(~tid);
  }
  __syncthreads();

  for (int kk = 2; kk <= 64; kk <<= 1) {
    for (int jj = kk >> 1; jj > 0; jj >>= 1) {
      int l = tid ^ jj;
      if (l > tid) {
        unsigned long long a = cand[tid], b = cand[l];
        bool up = ((tid & kk) == 0);
        if (up ? (a > b) : (a < b)) { cand[tid] = b; cand[l] = a; }
      }
      __syncthreads();
    }
  }

  if (tid < K3N * 8) {
    const int r = tid >> 3, o = tid & 7;
    unsigned long long c = cand[63 - r];
    unsigned int key = (unsigned int)(c >> 32);
    int idx = (int)(~(unsigned int)c);
    float val = tanhf(key2f(key) * inv_norm);
    const float* xr = yg + (size_t)idx * 32;
    float acc = b3[o];
#pragma unroll
    for (int f = 0; f < 32; ++f) acc += xr[f] * val * W3m[o * 32 + f];
    out[(size_t)(g * K3N + r) * 8 + o] = acc;
  }
}

// ---------------------------------------------------------------------------
// Launch
// ---------------------------------------------------------------------------
extern "C" void kernel_launch(void* const* d_in, const int* in_sizes, int n_in,
                              void* d_out, int out_size, void* d_ws, size_t ws_size,
                              hipStream_t stream) {
  const float* x    = (const float*)d_in[0];
  // d_in[1] edge_index, d_in[2] batch: unused by the reference computation
  const float* w1   = (const float*)d_in[3];
  const float* Wih  = (const float*)d_in[4];
  const float* bih  = (const float*)d_in[5];
  const float* bhh  = (const float*)d_in[6];
  const float* w2   = (const float*)d_in[7];
  const float* W2m  = (const float*)d_in[8];
  const float* b2   = (const float*)d_in[9];
  const float* w3   = (const float*)d_in[10];
  const float* W3m  = (const float*)d_in[11];
  const float* b3   = (const float*)d_in[12];
  float* out = (float*)d_out;

  float* ws  = (float*)d_ws;
  float* xp1 = ws;                            // M1 x 8
  float* y1  = xp1 + (size_t)M1 * 8;          // M1 x 64
  float* xp2 = y1  + (size_t)M1 * 64;         // M2 x 64
  float* y2  = xp2 + (size_t)M2 * 64;         // M2 x 32

  // pool1 needs 128KB keys + 1KB hist + ctrl + 4KB candidates of LDS
  const size_t smem1 = 131072 + 1024 + 32 + 4096;
  hipFuncSetAttribute((const void*)pool1_kernel,
                      hipFuncAttributeMaxDynamicSharedMemorySize, (int)smem1);

  pool1_kernel<<<BGRAPH, 1024, smem1, stream>>>(x, w1, xp1);
  rnn_gemm_wmma<<<(M1 / 16) * 4 / 8, 256, 0, stream>>>(xp1, Wih, bih, bhh, y1);
  pool2_kernel<<<BGRAPH, 256, 0, stream>>>(y1, w2, xp2);
  gemm2_wmma<<<(M2 / 16) * 2 / 8, 256, 0, stream>>>(xp2, W2m, b2, y2);
  pool3_final_kernel<<<BGRAPH, 64, 0, stream>>>(y2, w3, W3m, b3, out);
}